// ReprojectionMultiRigModelWithDepth_68839735820965
// MI455X (gfx1250) — compile-verified
//
#include <hip/hip_runtime.h>

#define BLOCK 256
#define EPSF 1e-6f

// Address-space-qualified pointer types for the gfx1250 async global->LDS builtins.
// Clang prints AS1 as '__device__ int *' and AS3 as '__shared__ int *' in HIP mode.
typedef __attribute__((address_space(1))) int* as1_i32p;
typedef __attribute__((address_space(3))) int* as3_i32p;

struct f3 { float x, y, z; };

__device__ __forceinline__ f3 cross3(f3 a, f3 b) {
    f3 r;
    r.x = a.y * b.z - a.z * b.y;
    r.y = a.z * b.x - a.x * b.z;
    r.z = a.x * b.y - a.y * b.x;
    return r;
}

// rotate v by quaternion (qx,qy,qz,qw):  v + qw*(2*qv x v) + qv x (2*qv x v)
__device__ __forceinline__ f3 quat_rot(float qx, float qy, float qz, float qw, f3 v) {
    f3 qv; qv.x = qx; qv.y = qy; qv.z = qz;
    f3 t = cross3(qv, v);
    t.x *= 2.0f; t.y *= 2.0f; t.z *= 2.0f;
    f3 c = cross3(qv, t);
    f3 r;
    r.x = v.x + qw * t.x + c.x;
    r.y = v.y + qw * t.y + c.y;
    r.z = v.z + qw * t.z + c.z;
    return r;
}

__global__ __launch_bounds__(BLOCK) void reproj_rig_depth_kernel(
    const float* __restrict__ points_2d,    // (N,2)
    const int*   __restrict__ cam_idx,      // (N,)
    const int*   __restrict__ grouping,     // (N,2) [group, member]
    const int*   __restrict__ pt_idx,       // (N,)
    const float* __restrict__ camera_pps,   // (128,2)
    const float* __restrict__ depths_ref,   // (N,)
    const float* __restrict__ intrs,        // (128,2)
    const float* __restrict__ points_3d,    // (1M,3)
    const float* __restrict__ ref_poses,    // (20000,7) [t(3), q(4)]
    const float* __restrict__ rel_poses,    // (16,7)
    float*       __restrict__ out,          // (N,3)
    int n)
{
    __shared__ float s_rel[16 * 7];      // 112 dwords
    __shared__ float s_cam[128 * 4];     // fx fy px py per cam, 512 dwords

    const int tid = threadIdx.x;

    // ---- Stage broadcast tables into LDS (CDNA5 async global->LDS path) ----
#if defined(__gfx1250__) && __has_builtin(__builtin_amdgcn_global_load_async_to_lds_b32)
    if (tid < 112) {
        __builtin_amdgcn_global_load_async_to_lds_b32(
            (as1_i32p)(rel_poses + tid),
            (as3_i32p)(s_rel + tid),
            0, 0);
    }
    for (int j = tid; j < 512; j += BLOCK) {
        const float* src;
        int dst;
        if (j < 256) { src = intrs + j;        dst = 4 * (j >> 1) + (j & 1); }
        else { int k = j - 256; src = camera_pps + k; dst = 4 * (k >> 1) + 2 + (k & 1); }
        __builtin_amdgcn_global_load_async_to_lds_b32(
            (as1_i32p)src,
            (as3_i32p)(s_cam + dst),
            0, 0);
    }
#if __has_builtin(__builtin_amdgcn_s_wait_asynccnt)
    __builtin_amdgcn_s_wait_asynccnt(0);
#else
    asm volatile("s_wait_asynccnt 0" ::: "memory");
#endif
#else
    if (tid < 112) s_rel[tid] = rel_poses[tid];
    for (int j = tid; j < 512; j += BLOCK) {
        if (j < 256) { s_cam[4 * (j >> 1) + (j & 1)] = intrs[j]; }
        else { int k = j - 256; s_cam[4 * (k >> 1) + 2 + (k & 1)] = camera_pps[k]; }
    }
#endif
    __syncthreads();

    const int stride = gridDim.x * BLOCK;
    for (int i = blockIdx.x * BLOCK + tid; i < n; i += stride) {
        // b64 loads for the paired index / uv streams
        const int2   gm = ((const int2*)grouping)[i];
        const float2 p2 = ((const float2*)points_2d)[i];
        const int g = gm.x;
        const int m = gm.y;
        const int c = cam_idx[i];
        const int p = pt_idx[i];

        // rel pose (LDS)
        const float* rel = &s_rel[m * 7];
        const float rtx = rel[0], rty = rel[1], rtz = rel[2];
        const float rqx = rel[3], rqy = rel[4], rqz = rel[5], rqw = rel[6];

        // ref pose (global gather, 28B row; L2-resident)
        const float* ref = ref_poses + (long long)g * 7;
        f3 ft; ft.x = ref[0]; ft.y = ref[1]; ft.z = ref[2];
        const float fqx = ref[3], fqy = ref[4], fqz = ref[5], fqw = ref[6];

        // pose.t = rel.t + R(rel.q) * ref.t
        f3 rt = quat_rot(rqx, rqy, rqz, rqw, ft);
        const float ptx = rtx + rt.x;
        const float pty = rty + rt.y;
        const float ptz = rtz + rt.z;

        // pose.q = rel.q * ref.q
        const float qw = rqw * fqw - (rqx * fqx + rqy * fqy + rqz * fqz);
        f3 v1; v1.x = rqx; v1.y = rqy; v1.z = rqz;
        f3 v2; v2.x = fqx; v2.y = fqy; v2.z = fqz;
        f3 cr = cross3(v1, v2);
        const float qx = rqw * fqx + fqw * rqx + cr.x;
        const float qy = rqw * fqy + fqw * rqy + cr.y;
        const float qz = rqw * fqz + fqw * rqz + cr.z;

        // p_cam = R(pose.q) * point + pose.t
        const float* P = points_3d + (long long)p * 3;
        f3 pt3; pt3.x = P[0]; pt3.y = P[1]; pt3.z = P[2];
        f3 pc = quat_rot(qx, qy, qz, qw, pt3);
        pc.x += ptx; pc.y += pty; pc.z += ptz;

        const float z    = pc.z;
        const float invz = 1.0f / z;                  // projection uses exact 1/z
        const float fx = s_cam[4 * c + 0];
        const float fy = s_cam[4 * c + 1];
        const float px = s_cam[4 * c + 2];
        const float py = s_cam[4 * c + 3];

        const float ru = fx * (pc.x * invz) + px - p2.x;
        const float rv = fy * (pc.y * invz) + py - p2.y;
        const float rd = 1.0f / (z + EPSF) - depths_ref[i];  // DEPTH_WEIGHT = 1.0

        float* o = out + (long long)i * 3;
        o[0] = ru;
        o[1] = rv;
        o[2] = rd;
    }
}

extern "C" void kernel_launch(void* const* d_in, const int* in_sizes, int n_in,
                              void* d_out, int out_size, void* d_ws, size_t ws_size,
                              hipStream_t stream) {
    (void)n_in; (void)out_size; (void)d_ws; (void)ws_size;

    // setup_inputs() dict order:
    const float* points_2d  = (const float*)d_in[0];
    const int*   cam_idx    = (const int*)  d_in[1];
    const int*   grouping   = (const int*)  d_in[2];
    const int*   pt_idx     = (const int*)  d_in[3];
    const float* camera_pps = (const float*)d_in[4];
    const float* depths_ref = (const float*)d_in[5];
    const float* intrs      = (const float*)d_in[6];
    const float* points_3d  = (const float*)d_in[7];
    const float* ref_poses  = (const float*)d_in[8];
    const float* rel_poses  = (const float*)d_in[9];

    const int n = in_sizes[1];  // camera_indices has N elements

    int blocks = (n + BLOCK - 1) / BLOCK;
    if (blocks > 4096) blocks = 4096;  // grid-stride; keeps LDS staging traffic tiny
    if (blocks < 1) blocks = 1;

    reproj_rig_depth_kernel<<<dim3(blocks), dim3(BLOCK), 0, stream>>>(
        points_2d, cam_idx, grouping, pt_idx, camera_pps, depths_ref,
        intrs, points_3d, ref_poses, rel_poses, (float*)d_out, n);
}